// SocialCircleLayer_42855183679744
// MI455X (gfx1250) — compile-verified
//
#include <hip/hip_runtime.h>
#include <math.h>

// SocialCircle fused kernel for gfx1250 (MI455X).
// Input order: trajs(unused), nei_trajs, w1_va, b1_va, w2_va, b2_va,
//              w1_dd, b1_dd, w2_dd, b2_dd.
// Output: (B, 8, 128) float32.

typedef __attribute__((ext_vector_type(2))) float v2f;
typedef __attribute__((ext_vector_type(8))) float v8f;

#define TWO_PI_F 6.28318530717958647692f

__global__ __launch_bounds__(256) void socialcircle_fused_kernel(
    const float* __restrict__ nei,    // (B, 512, 8, 2)
    const float* __restrict__ w1va,   // (1, 64)
    const float* __restrict__ b1va,   // (64)
    const float* __restrict__ w2va,   // (64, 64)
    const float* __restrict__ b2va,   // (64)
    const float* __restrict__ w1dd,   // (2, 64)
    const float* __restrict__ b1dd,   // (64)
    const float* __restrict__ w2dd,   // (64, 64)
    const float* __restrict__ b2dd,   // (64)
    float* __restrict__ out)          // (B, 8, 128)
{
    __shared__ float s_w2va[64 * 64];      // 16 KB
    __shared__ float s_w2dd[64 * 64];      // 16 KB
    __shared__ float s_w1va[64], s_b1va[64];
    __shared__ float s_w1dd[128], s_b1dd[64];
    __shared__ float s_b2va[64], s_b2dd[64];
    __shared__ float s_red[2][4][8];       // [local batch][vel,dis,dir,cnt][partition]
    __shared__ float s_feat[16][3];        // 16 rows x {vel, dis, dir}

    const int tid = threadIdx.x;

    // ---- Stage weights into LDS ----
    {
        const float4* sva = (const float4*)w2va;
        const float4* sdd = (const float4*)w2dd;
        float4* dva = (float4*)s_w2va;
        float4* ddd = (float4*)s_w2dd;
#pragma unroll
        for (int i = 0; i < 4; ++i) {      // 1024 float4 per matrix / 256 threads
            dva[tid + 256 * i] = sva[tid + 256 * i];
            ddd[tid + 256 * i] = sdd[tid + 256 * i];
        }
        if (tid < 64) {
            s_w1va[tid] = w1va[tid];
            s_b1va[tid] = b1va[tid];
            s_b1dd[tid] = b1dd[tid];
            s_b2va[tid] = b2va[tid];
            s_b2dd[tid] = b2dd[tid];
            ((float*)s_red)[tid] = 0.0f;   // zero the 64 reduction slots
        } else if (tid < 192) {
            s_w1dd[tid - 64] = w1dd[tid - 64];
        }
    }
    __syncthreads();

    // ---- Reduction over 2 batches x 512 neighbors (1024 neighbors/block) ----
    // Each neighbor = 16 contiguous floats (T=8, xy). One pass, fully coalesced.
    const long long nbase = (long long)blockIdx.x * 2 * 512 * 16;
#pragma unroll
    for (int i = 0; i < 4; ++i) {
        const int idx = tid + 256 * i;     // 0..1023
        const int lb  = idx >> 9;          // local batch 0/1
        const float4* p = (const float4*)(nei + nbase + (long long)idx * 16);
        float4 f0 = p[0];
        float4 f1 = p[1];
        float4 f2 = p[2];
        float4 f3 = p[3];
        float s = f0.x + f0.y + f0.z + f0.w
                + f1.x + f1.y + f1.z + f1.w
                + f2.x + f2.y + f2.z + f2.w
                + f3.x + f3.y + f3.z + f3.w;
        if (s != 0.0f) {                   // nei_mask: all-zero trajectory excluded
            const float px = f3.z, py = f3.w;          // T=7 position
            const float vx = px - f0.x, vy = py - f0.y; // last - first
            const float fvel = sqrtf(vx * vx + vy * vy);
            const float fdis = sqrtf(px * px + py * py);
            float fdir = atan2f(px, py);               // arctan2(x, y) per reference
            if (fdir < 0.0f) fdir += TWO_PI_F;
            int part = (int)(fdir * (8.0f / TWO_PI_F));
            part = part > 7 ? 7 : part;
            atomicAdd(&s_red[lb][0][part], fvel);
            atomicAdd(&s_red[lb][1][part], fdis);
            atomicAdd(&s_red[lb][2][part], fdir);
            atomicAdd(&s_red[lb][3][part], 1.0f);
        }
    }
    __syncthreads();

    if (tid < 16) {
        const int lb = tid >> 3, p = tid & 7;
        const float n = s_red[lb][3][p] + 1e-4f;
        s_feat[tid][0] = s_red[lb][0][p] / n;
        s_feat[tid][1] = s_red[lb][1][p] / n;
        s_feat[tid][2] = s_red[lb][2][p] / n;
    }
    __syncthreads();

    // ---- MLP phase: 16 rows x 64 hidden, two MLPs, via V_WMMA_F32_16X16X4_F32 ----
    // Waves 0-3: "va" MLP (cols 0..63); waves 4-7: "dd" MLP (cols 64..127).
    const int wave = tid >> 5;
    const int lane = tid & 31;
    const int mlp  = wave >> 2;            // 0 = vel/acc branch, 1 = dis/dir branch
    const int n0   = (wave & 3) * 16;      // N-tile base within this MLP's 64 cols
    const int m    = lane & 15;            // A-matrix row (M) for this lane
    const int hi   = lane >> 4;            // 0: K={k,k+1}, 1: K={k+2,k+3}
    const int nn   = n0 + (lane & 15);     // B/C/D column for this lane

    const float  x0  = mlp ? s_feat[m][1] : s_feat[m][0];
    const float  x1  = mlp ? s_feat[m][2] : 0.0f;
    const float* wp0 = mlp ? s_w1dd       : s_w1va;
    const float* wp1 = mlp ? s_w1dd + 64  : s_w1va;  // dead (x1==0) for va branch
    const float* bp1 = mlp ? s_b1dd       : s_b1va;
    const float* w2p = mlp ? s_w2dd       : s_w2va;
    const float* b2p = mlp ? s_b2dd       : s_b2va;
    const int colbase = mlp ? 64 : 0;

    v8f acc = {};
#pragma unroll
    for (int k0 = 0; k0 < 64; k0 += 4) {
        const int j = k0 + 2 * hi;
        // Hidden layer computed directly in A-matrix register layout:
        const float h0 = x0 * wp0[j]     + x1 * wp1[j]     + bp1[j];
        const float h1 = x0 * wp0[j + 1] + x1 * wp1[j + 1] + bp1[j + 1];
        v2f a, b;
        a.x = h0 > 0.0f ? h0 : 0.0f;       // relu
        a.y = h1 > 0.0f ? h1 : 0.0f;
        // B-matrix (4x16): row K=j / j+1 at column nn
        b.x = w2p[j * 64 + nn];
        b.y = w2p[(j + 1) * 64 + nn];
        acc = __builtin_amdgcn_wmma_f32_16x16x4_f32(
            /*neg_a=*/false, a, /*neg_b=*/false, b,
            /*c_mod=*/(short)0, acc, /*reuse_a=*/false, /*reuse_b=*/false);
    }

    // ---- Epilogue: D layout -> tanh(acc + b2) -> global ----
    const float bias = b2p[nn];
    const long long rowbase = (long long)blockIdx.x * 16;  // rows = b*8 + p
#pragma unroll
    for (int r = 0; r < 8; ++r) {
        const int mm = r + 8 * hi;         // C/D: lanes 16-31 hold M = r+8
        const float v = tanhf(acc[r] + bias);
        out[(rowbase + mm) * 128 + colbase + nn] = v;
    }
}

extern "C" void kernel_launch(void* const* d_in, const int* in_sizes, int n_in,
                              void* d_out, int out_size, void* d_ws, size_t ws_size,
                              hipStream_t stream) {
    (void)n_in; (void)out_size; (void)d_ws; (void)ws_size;
    // Input order from setup_inputs(): trajs(0, unused), nei_trajs(1),
    // w1_va(2), b1_va(3), w2_va(4), b2_va(5), w1_dd(6), b1_dd(7), w2_dd(8), b2_dd(9)
    const float* nei  = (const float*)d_in[1];
    const float* w1va = (const float*)d_in[2];
    const float* b1va = (const float*)d_in[3];
    const float* w2va = (const float*)d_in[4];
    const float* b2va = (const float*)d_in[5];
    const float* w1dd = (const float*)d_in[6];
    const float* b1dd = (const float*)d_in[7];
    const float* w2dd = (const float*)d_in[8];
    const float* b2dd = (const float*)d_in[9];
    float* out = (float*)d_out;

    const int B = in_sizes[0] / 16;        // trajs is (B, 8, 2)
    const int grid = B / 2;                // 2 batches (16 output rows) per block

    hipLaunchKernelGGL(socialcircle_fused_kernel, dim3(grid), dim3(256), 0, stream,
                       nei, w1va, b1va, w2va, b2va, w1dd, b1dd, w2dd, b2dd, out);
}